// PSROIPool_73873437491276
// MI455X (gfx1250) — compile-verified
//
#include <hip/hip_runtime.h>
#include <stdint.h>

// PS-ROI pooling, direct formulation (no SAT): each output bin is the mean of
// a small (<= ~6x6) rectangle of one channel. features fits in MI455X's 192MB
// L2, so the kernel is bound by ~67MB cold HBM read + 4.2MB write (~3us).
//
// CDNA5 path: per-block ROI descriptor is staged into LDS with the async
// global->LDS DMA (ASYNCcnt) and synchronized with s_wait_asynccnt.

constexpr int   kPooled   = 7;
constexpr int   kOutDim   = 21;
constexpr int   kChannels = kOutDim * kPooled * kPooled;  // 1029
constexpr int   kH        = 64;
constexpr int   kW        = 64;
constexpr float kScale    = 0.0625f;
constexpr int   kBins     = kOutDim * kPooled * kPooled;  // 1029 outputs / ROI

__global__ __launch_bounds__(256) void psroi_pool_kernel(
    const float* __restrict__ features,
    const float* __restrict__ rois,
    float* __restrict__ out) {
  __shared__ float s_roi[8];                       // staged ROI row (5 used)
  __shared__ int s_hs[kPooled], s_he[kPooled];
  __shared__ int s_ws[kPooled], s_we[kPooled];
  __shared__ int s_b;

  const int roi = blockIdx.x;
  const int tid = threadIdx.x;

  // ---- gfx1250 async global->LDS stage of this block's ROI row (5 floats).
  // Lanes 0..4 of wave 0 each DMA one dword; EXEC from the branch gates lanes.
  if (tid < 5) {
    const float* gsrc = rois + roi * 5 + tid;
    // Truncating a flat-cast LDS pointer yields the 32-bit LDS byte offset
    // (LDS aperture occupies addr[63:32]; addr[31:0] is the offset).
    uint32_t lds_off = (uint32_t)(uintptr_t)(&s_roi[tid]);
    asm volatile("global_load_async_to_lds_b32 %0, %1, off"
                 :
                 : "v"(lds_off), "v"(gsrc)
                 : "memory");
  }
  asm volatile("s_wait_asynccnt 0x0" ::: "memory");  // ASYNCcnt == 0
  __syncthreads();

  // ---- Precompute per-ph / per-pw bin bounds (reference numerics).
  if (tid < 2 * kPooled) {
    const float rsw = rintf(s_roi[1]) * kScale;          // round-half-even
    const float rsh = rintf(s_roi[2]) * kScale;
    const float rew = rintf(s_roi[3] + 1.0f) * kScale;
    const float reh = rintf(s_roi[4] + 1.0f) * kScale;
    const float rw  = fmaxf(rew - rsw, 0.1f);
    const float rh  = fmaxf(reh - rsh, 0.1f);
    const float bw  = rw / (float)kPooled;               // true fp32 divide
    const float bh  = rh / (float)kPooled;
    const int   p   = tid % kPooled;
    const float fp  = (float)p;
    if (tid < kPooled) {
      // no fma-contraction: match jnp's separate mul + add before floor/ceil
      s_hs[p] = (int)fminf(fmaxf(floorf(__fadd_rn(__fmul_rn(fp, bh), rsh)),
                                 0.0f), (float)kH);
      s_he[p] = (int)fminf(fmaxf(ceilf(__fadd_rn(__fmul_rn(fp + 1.0f, bh), rsh)),
                                 0.0f), (float)kH);
    } else {
      s_ws[p] = (int)fminf(fmaxf(floorf(__fadd_rn(__fmul_rn(fp, bw), rsw)),
                                 0.0f), (float)kW);
      s_we[p] = (int)fminf(fmaxf(ceilf(__fadd_rn(__fmul_rn(fp + 1.0f, bw), rsw)),
                                 0.0f), (float)kW);
    }
  }
  if (tid == 2 * kPooled) s_b = (int)s_roi[0];
  __syncthreads();

  const float* fbase = features + (size_t)s_b * kChannels * kH * kW;

  // ---- 1029 outputs per ROI, 256 threads -> <=5 iterations each.
  for (int j = tid; j < kBins; j += 256) {
    const int d  = j / (kPooled * kPooled);
    const int r  = j - d * (kPooled * kPooled);
    const int ph = r / kPooled;
    const int pw = r - ph * kPooled;
    // POOLED == GROUP_SIZE == 7, so gh == ph and gw == pw.
    const int chan = (d * kPooled + ph) * kPooled + pw;

    const int hs = s_hs[ph], he = s_he[ph];
    const int ws = s_ws[pw], we = s_we[pw];

    const float* cbase = fbase + (size_t)chan * (kH * kW);
    float sum = 0.0f;
    for (int h = hs; h < he; ++h) {
      const float* row = cbase + h * kW + ws;
      for (int w = 0; w < we - ws; ++w) sum += row[w];
    }
    const int area = (he - hs) * (we - ws);
    out[(size_t)roi * kBins + j] = (area <= 0) ? 0.0f : sum / (float)area;
  }
}

extern "C" void kernel_launch(void* const* d_in, const int* in_sizes, int n_in,
                              void* d_out, int out_size, void* d_ws, size_t ws_size,
                              hipStream_t stream) {
  const float* features = (const float*)d_in[0];
  const float* rois     = (const float*)d_in[1];
  float*       out      = (float*)d_out;

  const int num_rois = in_sizes[1] / 5;  // rois is (N, 5)
  psroi_pool_kernel<<<dim3(num_rois), dim3(256), 0, stream>>>(features, rois, out);
}